// Transformer_CRF_48017734369683
// MI455X (gfx1250) — compile-verified
//
#include <hip/hip_runtime.h>
#include <cstdint>
#include <cstddef>

// ---------------------------------------------------------------------------
// CRF (forward + Viterbi + backtrack + loss) for B=256, T=1024, L=128.
//
// Forward recurrence is computed in the exp-domain with per-step, per-batch
// max-normalization so each step is a dense fp32 GEMM:
//     D[b,i] = sum_j exp(alpha[b,j]-m_b) * exp(T[i,j])
//     alpha'[b,i] = log(D[b,i]) + m_b + feat[b,t,i]
// executed with V_WMMA_F32_16X16X4_F32 (fp32 accumulate == FMA-exact).
// Viterbi (max-plus) stays on VALU with T[i,:] held in VGPRs.
// ---------------------------------------------------------------------------

#define NB 256
#define NT 1024
#define NL 128
#define NEG_INF_F (-10000.0f)

typedef __attribute__((ext_vector_type(2))) float v2f;
typedef __attribute__((ext_vector_type(8))) float v8f;

// ---------------------------------------------------------------------------
// Prep: ET[j*L+i] = exp(T[i][j])  (B-matrix for WMMA, K=j rows, N=i cols)
//       Ttr[j*L+i] = T[i][j]      (coalesced column fetch for Viterbi)
// ---------------------------------------------------------------------------
__global__ __launch_bounds__(256) void crf_prep(const float* __restrict__ trans,
                                                float* __restrict__ ET,
                                                float* __restrict__ Ttr) {
  int idx = blockIdx.x * 256 + threadIdx.x;  // 0..16383
  int i = idx >> 7;
  int j = idx & 127;
  float v = trans[i * NL + j];
  ET[j * NL + i] = __expf(v);
  Ttr[j * NL + i] = v;
}

// ---------------------------------------------------------------------------
// Forward: 16 blocks x 256 threads. Block = 16 batches. Wave w owns labels
// [16w,16w+16). B-panel (128x16 slice of ET) lives in 64 VGPRs per lane.
// Per step: 32 x v_wmma_f32_16x16x4_f32 accumulation over K=128.
// ---------------------------------------------------------------------------
__global__ __launch_bounds__(256) void crf_forward(const float* __restrict__ feats,
                                                   const float* __restrict__ ET,
                                                   float* __restrict__ fwd_score) {
  constexpr int LP = 132;  // padded row stride to spread LDS banks
  __shared__ __align__(16) float alpha[16 * LP];
  __shared__ __align__(16) float Abuf[16 * LP];
  __shared__ float pmax[16 * 16];
  __shared__ float mrow[16];

  const int tid = threadIdx.x;
  const int wave = tid >> 5;    // 0..7 -> N tile
  const int lane = tid & 31;
  const int half = lane >> 4;   // 0 | 1 (K split per WMMA layout)
  const int l15 = lane & 15;
  const int b0 = blockIdx.x * 16;

  // ---- load B-panel into registers: Bmat[k][n] = ET[k*L + (16w+l15)]
  // layout: vgpr.x -> K = 4kk + 2*half, vgpr.y -> K+1 (mirrors A layout)
  v2f bp[32];
#pragma unroll
  for (int kk = 0; kk < 32; ++kk) {
    const float* p = ET + (size_t)(4 * kk + 2 * half) * NL + wave * 16 + l15;
    bp[kk].x = p[0];
    bp[kk].y = p[NL];
  }

  // ---- init alpha_0, A_0 = exp(alpha_0 - 0), m = 0
  for (int idx = tid; idx < 16 * NL; idx += 256) {
    int b = idx >> 7, i = idx & 127;
    alpha[b * LP + i] = (i == 0) ? 0.0f : NEG_INF_F;
    Abuf[b * LP + i] = (i == 0) ? 1.0f : 0.0f;
  }
  if (tid < 16) mrow[tid] = 0.0f;
  __syncthreads();

  for (int t = 1; t < NT; ++t) {
    // prefetch next step's emission tile (global_prefetch_b8 path)
    if (t + 1 < NT) {
      const float* nf =
          feats + ((size_t)(b0 + (tid >> 4)) * NT + (t + 1)) * NL + (tid & 15) * 8;
      __builtin_prefetch(nf, 0, 1);
    }

    // ---- D = A x Bpanel  (A: 16 batches x K=128, fp32 WMMA chain)
    v8f c = {};
#pragma unroll
    for (int kk = 0; kk < 32; ++kk) {
      // A layout: lane m=l15 row; .x -> K=4kk+2*half, .y -> K+1
      v2f a = *(const v2f*)&Abuf[l15 * LP + 4 * kk + 2 * half];
      c = __builtin_amdgcn_wmma_f32_16x16x4_f32(false, a, false, bp[kk],
                                                (short)0, c, false, false);
    }

    // ---- alpha' = log(D) + m_b + feat ; C/D layout: vgpr v -> batch v+8*half
#pragma unroll
    for (int v = 0; v < 8; ++v) {
      int b = v + 8 * half;
      int i = wave * 16 + l15;
      float d = fmaxf(c[v], 1e-37f);
      float fe = feats[((size_t)(b0 + b) * NT + t) * NL + i];
      alpha[b * LP + i] = __logf(d) + mrow[b] + fe;
    }
    __syncthreads();

    // ---- m_b = max_i alpha'  (two-level reduction)
    {
      int b = tid >> 4, s = tid & 15;
      float mv = -3.4e38f;
#pragma unroll
      for (int q = 0; q < 8; ++q) mv = fmaxf(mv, alpha[b * LP + s * 8 + q]);
      pmax[b * 16 + s] = mv;
    }
    __syncthreads();
    if (tid < 16) {
      float mv = pmax[tid * 16];
#pragma unroll
      for (int s = 1; s < 16; ++s) mv = fmaxf(mv, pmax[tid * 16 + s]);
      mrow[tid] = mv;
    }
    __syncthreads();

    // ---- A = exp(alpha' - m_b)
    for (int idx = tid; idx < 16 * NL; idx += 256) {
      int b = idx >> 7, i = idx & 127;
      Abuf[b * LP + i] = __expf(alpha[b * LP + i] - mrow[b]);
    }
    __syncthreads();
  }

  // forward_score[b] = m_b + log(sum_i exp(alpha-m_b))  (Abuf holds exps)
  {
    int b = tid >> 4, s = tid & 15;
    float sv = 0.0f;
#pragma unroll
    for (int q = 0; q < 8; ++q) sv += Abuf[b * LP + s * 8 + q];
    pmax[b * 16 + s] = sv;
  }
  __syncthreads();
  if (tid < 16) {
    float sv = 0.0f;
#pragma unroll
    for (int s = 0; s < 16; ++s) sv += pmax[tid * 16 + s];
    fwd_score[b0 + tid] = mrow[tid] + __logf(sv);
  }
}

// ---------------------------------------------------------------------------
// Viterbi: 64 blocks x 256 threads, 4 batches/block. Thread owns label i =
// tid&127 and two batches; T[i,:] held in 128 VGPRs; delta ping-pongs in LDS.
// psi stored as u8 (label < 256).
// ---------------------------------------------------------------------------
__global__ __launch_bounds__(256) void crf_viterbi(const float* __restrict__ feats,
                                                   const float* __restrict__ Ttr,
                                                   uint8_t* __restrict__ psi,
                                                   int* __restrict__ last,
                                                   float* __restrict__ maxll) {
  constexpr int LPV = 132;
  __shared__ float dbuf[2][4][LPV];

  const int tid = threadIdx.x;
  const int i = tid & 127;
  const int g = tid >> 7;  // 0..1
  const int b0 = blockIdx.x * 4;

  float treg[NL];
#pragma unroll
  for (int j = 0; j < NL; ++j) treg[j] = Ttr[j * NL + i];  // = T[i][j]

  if (g == 0) {
#pragma unroll
    for (int b = 0; b < 4; ++b) dbuf[0][b][i] = (i == 0) ? 0.0f : NEG_INF_F;
  }
  __syncthreads();

  for (int t = 1; t < NT; ++t) {
    const int pp = (t - 1) & 1, cc = t & 1;
#pragma unroll
    for (int u = 0; u < 2; ++u) {
      const int b = g * 2 + u;
      float best = -3.0e38f;
      int bi = 0;
#pragma unroll
      for (int j = 0; j < NL; ++j) {
        float s = treg[j] + dbuf[pp][b][j];
        bi = (s > best) ? j : bi;
        best = fmaxf(best, s);
      }
      float fe = feats[((size_t)(b0 + b) * NT + t) * NL + i];
      dbuf[cc][b][i] = best + fe;
      psi[((size_t)(t - 1) * NB + (b0 + b)) * NL + i] = (uint8_t)bi;
    }
    __syncthreads();
  }

  if (tid < 4) {
    const int cc = (NT - 1) & 1;
    float best = -3.0e38f;
    int bi = 0;
    for (int j = 0; j < NL; ++j) {
      float v = dbuf[cc][tid][j];
      if (v > best) { best = v; bi = j; }
    }
    maxll[b0 + tid] = best;
    last[b0 + tid] = bi;
  }
}

// ---------------------------------------------------------------------------
// Backtrack: 1 block, thread b walks its psi chain (serial dependence).
// ---------------------------------------------------------------------------
__global__ __launch_bounds__(256) void crf_backtrack(const uint8_t* __restrict__ psi,
                                                     const int* __restrict__ last,
                                                     float* __restrict__ path) {
  int b = threadIdx.x;
  int p = last[b];
  path[(size_t)b * NT + (NT - 1)] = (float)p;
  for (int s = NT - 2; s >= 0; --s) {
    p = psi[((size_t)s * NB + b) * NL + p];
    path[(size_t)b * NT + s] = (float)p;
  }
}

// ---------------------------------------------------------------------------
// Gold score + loss: loss = mean_b(forward_score[b] - gold[b])
// ---------------------------------------------------------------------------
__global__ __launch_bounds__(256) void crf_gold_loss(const float* __restrict__ feats,
                                                     const int* __restrict__ labels,
                                                     const float* __restrict__ trans,
                                                     const float* __restrict__ fwd,
                                                     float* __restrict__ loss) {
  __shared__ float red[256];
  int b = threadIdx.x;
  float gsum = 0.0f;
  int yp = labels[b * NT];
  for (int t = 1; t < NT; ++t) {
    int yt = labels[b * NT + t];
    gsum += trans[yt * NL + yp] + feats[((size_t)b * NT + t) * NL + yt];
    yp = yt;
  }
  red[b] = fwd[b] - gsum;
  __syncthreads();
  for (int off = 128; off > 0; off >>= 1) {
    if (b < off) red[b] += red[b + off];
    __syncthreads();
  }
  if (b == 0) loss[0] = red[0] / (float)NB;
}

// ---------------------------------------------------------------------------
extern "C" void kernel_launch(void* const* d_in, const int* in_sizes, int n_in,
                              void* d_out, int out_size, void* d_ws, size_t ws_size,
                              hipStream_t stream) {
  (void)in_sizes; (void)n_in; (void)out_size; (void)ws_size;

  const float* feats = (const float*)d_in[0];   // (256,1024,128) f32
  const int* labels = (const int*)d_in[1];      // (256,1024) i32
  const float* trans = (const float*)d_in[2];   // (128,128) f32

  float* out = (float*)d_out;
  float* path = out;                 // 256*1024
  float* maxll = out + NB * NT;      // 256
  float* lossp = maxll + NB;         // 1

  char* ws = (char*)d_ws;
  float* ET = (float*)(ws);                 //  64 KB: exp(T) transposed
  float* Ttr = (float*)(ws + 65536);        //  64 KB: T transposed
  float* fwd = (float*)(ws + 131072);       //   1 KB: forward scores
  int* last = (int*)(ws + 132096);          //   1 KB: argmax at T-1
  uint8_t* psi = (uint8_t*)(ws + 133120);   // 33.5 MB: backpointers (u8)

  crf_prep<<<64, 256, 0, stream>>>(trans, ET, Ttr);
  crf_forward<<<16, 256, 0, stream>>>(feats, ET, fwd);
  crf_viterbi<<<64, 256, 0, stream>>>(feats, Ttr, psi, last, maxll);
  crf_backtrack<<<1, 256, 0, stream>>>(psi, last, path);
  crf_gold_loss<<<1, 256, 0, stream>>>(feats, labels, trans, fwd, lossp);
}